// ELADecoder_56573309223004
// MI455X (gfx1250) — compile-verified
//
#include <hip/hip_runtime.h>
#include <hip/hip_bf16.h>

typedef __attribute__((ext_vector_type(2))) float v2f;
typedef __attribute__((ext_vector_type(8))) float v8f;

#define D_MODEL 256
#define N_HEADS 8
#define D_HEAD  32
#define BS      8
#define LQ      300
#define LV      13294   // 100*100 + 50*50 + 25*25 + 13*13

#define WMMA_F32(a, b, c) \
    __builtin_amdgcn_wmma_f32_16x16x4_f32(false, (a), false, (b), (short)0, (c), false, false)

// ---------------------------------------------------------------------------
// WMMA fp32 GEMM:  C[M,N] = A[M,K] * W[N,K]^T + bias[N]
// One wave computes a 16(M) x 64(N) tile: one A fragment is reused across
// 4 chained V_WMMA_F32_16X16X4_F32 ops per k-step (5 loads : 4 WMMAs).
// The k-loop is 2-stage software-pipelined in source: fragments for step
// k+1 are issued before the WMMAs of step k, so the compiler can overlap
// global_load_b64 latency with matrix ops (partial s_wait_loadcnt).
// Requires M%16==0, N%64==0, K%8==0 (true for all GEMMs here).
// Wave-uniform early exit keeps EXEC all-ones as WMMA requires.
//
// Fragment layouts (ISA 7.12.2, 32-bit A 16x4 / B 4x16 / C 16x16):
//   A: lanes 0-15 -> M=lane, VGPR0=K+0, VGPR1=K+1 ; lanes 16-31 -> K+2,K+3
//   B: lanes 0-15 -> N=lane, VGPR0=K+0, VGPR1=K+1 ; lanes 16-31 -> K+2,K+3
//   C: VGPR r -> (M=r, N=lane) for lanes 0-15, (M=r+8, N=lane-16) for 16-31
// ---------------------------------------------------------------------------
__global__ void wmma_gemm_bias(const float* __restrict__ A,
                               const float* __restrict__ W,
                               const float* __restrict__ bias,
                               float* __restrict__ C,
                               int M, int N, int K)
{
    const int wavesPerBlock = blockDim.x >> 5;
    const int wave = blockIdx.x * wavesPerBlock + (threadIdx.x >> 5);
    const int ntiles = N >> 6;                 // 64-wide N tiles
    const int tm = (wave / ntiles) << 4;
    const int tn = (wave % ntiles) << 6;
    if (tm >= M) return;                       // wave-uniform -> EXEC stays full

    const int lane = threadIdx.x & 31;
    const int half = lane >> 4;                // 0: lanes 0-15, 1: lanes 16-31
    const int l16  = lane & 15;

    const float* __restrict__ arow = A + (size_t)(tm + l16) * K + half * 2;
    const float* __restrict__ w0   = W + (size_t)(tn + l16) * K + half * 2;
    const float* __restrict__ w1   = w0 + (size_t)16 * K;
    const float* __restrict__ w2   = w0 + (size_t)32 * K;
    const float* __restrict__ w3   = w0 + (size_t)48 * K;

    v8f acc0 = {}, acc1 = {}, acc2 = {}, acc3 = {};

    // prologue: fragments for k = 0
    v2f a  = *(const v2f*)(arow);
    v2f b0 = *(const v2f*)(w0);
    v2f b1 = *(const v2f*)(w1);
    v2f b2 = *(const v2f*)(w2);
    v2f b3 = *(const v2f*)(w3);

#pragma unroll 2
    for (int k = 4; k < K; k += 4) {
        // issue next iteration's loads before consuming current fragments
        v2f an  = *(const v2f*)(arow + k);
        v2f bn0 = *(const v2f*)(w0 + k);
        v2f bn1 = *(const v2f*)(w1 + k);
        v2f bn2 = *(const v2f*)(w2 + k);
        v2f bn3 = *(const v2f*)(w3 + k);

        acc0 = WMMA_F32(a, b0, acc0);
        acc1 = WMMA_F32(a, b1, acc1);
        acc2 = WMMA_F32(a, b2, acc2);
        acc3 = WMMA_F32(a, b3, acc3);

        a = an; b0 = bn0; b1 = bn1; b2 = bn2; b3 = bn3;
    }
    // epilogue: last k-step
    acc0 = WMMA_F32(a, b0, acc0);
    acc1 = WMMA_F32(a, b1, acc1);
    acc2 = WMMA_F32(a, b2, acc2);
    acc3 = WMMA_F32(a, b3, acc3);

    const float bv0 = bias[tn + l16];
    const float bv1 = bias[tn + 16 + l16];
    const float bv2 = bias[tn + 32 + l16];
    const float bv3 = bias[tn + 48 + l16];
#pragma unroll
    for (int r = 0; r < 8; ++r) {
        const size_t row = (size_t)(tm + r + half * 8) * N;
        C[row + tn      + l16] = acc0[r] + bv0;
        C[row + tn + 16 + l16] = acc1[r] + bv1;
        C[row + tn + 32 + l16] = acc2[r] + bv2;
        C[row + tn + 48 + l16] = acc3[r] + bv3;
    }
}

// ---------------------------------------------------------------------------
// In-place softmax over contiguous groups of 16 floats (L*P per head).
// Raw attn-logit layout [2400,128] makes (row,h) groups contiguous.
// ---------------------------------------------------------------------------
__global__ void softmax16_inplace(float* __restrict__ aw, int ngroups)
{
    const int t = blockIdx.x * blockDim.x + threadIdx.x;
    if (t >= ngroups) return;
    float* p = aw + (size_t)t * 16;
    float vals[16];
    float m = -3.402823466e+38f;
#pragma unroll
    for (int i = 0; i < 16; ++i) { vals[i] = p[i]; m = fmaxf(m, vals[i]); }
    float s = 0.f;
#pragma unroll
    for (int i = 0; i < 16; ++i) { vals[i] = __expf(vals[i] - m); s += vals[i]; }
    const float inv = 1.f / s;
#pragma unroll
    for (int i = 0; i < 16; ++i) p[i] = vals[i] * inv;
}

// ---------------------------------------------------------------------------
// Deformable bilinear sampling + attention-weighted blend.
// One wave per (b, q, h); lane = channel d within the head. Each corner
// gather reads 32 consecutive floats (128B) across the wave -> coalesced.
// Writes mid[b*LQ+q][h*32+d]  (the reference's transposed-back layout).
// ---------------------------------------------------------------------------
__device__ __forceinline__ float corner(const float* __restrict__ vl,
                                        int x, int y, int W, int H)
{
    if (x < 0 || x >= W || y < 0 || y >= H) return 0.f;
    return vl[(size_t)(y * W + x) * D_MODEL];
}

__global__ void ms_deform_sample(const float* __restrict__ vproj, // [BS,LV,256]
                                 const float* __restrict__ off,   // [BS*LQ,256]
                                 const float* __restrict__ aw,    // [BS*LQ,128] softmaxed
                                 const float* __restrict__ bbox,  // [BS,LQ,4,2]
                                 float* __restrict__ mid)         // [BS*LQ,256]
{
    const int wave = blockIdx.x * (blockDim.x >> 5) + (threadIdx.x >> 5);
    const int total = BS * LQ * N_HEADS;
    if (wave >= total) return;

    const int h   = wave % N_HEADS;
    const int row = wave / N_HEADS;          // b*LQ + q
    const int b   = row / LQ;
    const int lane = threadIdx.x & 31;       // channel d

    const int Hs[4]     = {100, 50, 25, 13};
    const int Wsz[4]    = {100, 50, 25, 13};
    const int starts[4] = {0, 10000, 12500, 13125};

    const float* __restrict__ offrow = off  + (size_t)row * D_MODEL + h * 32;
    const float* __restrict__ awrow  = aw   + (size_t)row * 128     + h * 16;
    const float* __restrict__ bb     = bbox + (size_t)row * 8;

    float acc = 0.f;
#pragma unroll
    for (int l = 0; l < 4; ++l) {
        const int H = Hs[l], W = Wsz[l];
        const float fW = (float)W, fH = (float)H;
        const float* __restrict__ vl =
            vproj + ((size_t)b * LV + starts[l]) * D_MODEL + h * D_HEAD + lane;
        const float rx = bb[l * 2 + 0];
        const float ry = bb[l * 2 + 1];
#pragma unroll
        for (int p = 0; p < 4; ++p) {
            const float ox = offrow[(l * 4 + p) * 2 + 0];
            const float oy = offrow[(l * 4 + p) * 2 + 1];
            // loc = ref + off/norm ; pixel coord = loc*size - 0.5
            const float px = (rx + ox / fW) * fW - 0.5f;
            const float py = (ry + oy / fH) * fH - 0.5f;
            const float x0f = floorf(px), y0f = floorf(py);
            const float wx = px - x0f,  wy = py - y0f;
            const int x0 = (int)x0f, y0 = (int)y0f;

            const float g00 = corner(vl, x0,     y0,     W, H);
            const float g01 = corner(vl, x0 + 1, y0,     W, H);
            const float g10 = corner(vl, x0,     y0 + 1, W, H);
            const float g11 = corner(vl, x0 + 1, y0 + 1, W, H);

            const float samp = g00 * (1.f - wx) * (1.f - wy)
                             + g01 * wx         * (1.f - wy)
                             + g10 * (1.f - wx) * wy
                             + g11 * wx         * wy;
            acc += awrow[l * 4 + p] * samp;
        }
    }
    mid[(size_t)row * D_MODEL + h * D_HEAD + lane] = acc;
}

// ---------------------------------------------------------------------------
// setup_inputs order:
//  0 query [8,300,256]      1 refer_bbox [8,300,4,2]   2 value [8,13294,256]
//  3 W_off [256,256]        4 b_off [256]              5 W_attn [128,256]
//  6 b_attn [128]           7 W_val [256,256]          8 b_val [256]
//  9 W_out [256,256]       10 b_out [256]
// out: [8,300,256] fp32
// ---------------------------------------------------------------------------
extern "C" void kernel_launch(void* const* d_in, const int* in_sizes, int n_in,
                              void* d_out, int out_size, void* d_ws, size_t ws_size,
                              hipStream_t stream)
{
    const float* query  = (const float*)d_in[0];
    const float* bbox   = (const float*)d_in[1];
    const float* value  = (const float*)d_in[2];
    const float* W_off  = (const float*)d_in[3];
    const float* b_off  = (const float*)d_in[4];
    const float* W_attn = (const float*)d_in[5];
    const float* b_attn = (const float*)d_in[6];
    const float* W_val  = (const float*)d_in[7];
    const float* b_val  = (const float*)d_in[8];
    const float* W_out  = (const float*)d_in[9];
    const float* b_out  = (const float*)d_in[10];
    float* out = (float*)d_out;

    // Workspace partition (floats): ~115 MB total
    float* ws     = (float*)d_ws;
    float* v_proj = ws;                                       // BS*LV*256
    float* off    = v_proj + (size_t)BS * LV * D_MODEL;       // 2400*256
    float* aw     = off    + (size_t)BS * LQ * D_MODEL;       // 2400*128
    float* mid    = aw     + (size_t)BS * LQ * 128;           // 2400*256

    const int MQ = BS * LQ;        // 2400 (= 150 * 16)
    const int MV = BS * LV;        // 106352 (= 6647 * 16)

    // 1) value projection: dominant GEMM, 13.9 GFLOP fp32 on the matrix pipe
    {
        const int waves = (MV / 16) * (D_MODEL / 64);
        wmma_gemm_bias<<<dim3((waves + 7) / 8), 256, 0, stream>>>(
            value, W_val, b_val, v_proj, MV, D_MODEL, D_MODEL);
    }
    // 2) sampling offsets: query @ W_off^T + b_off -> [2400,256]
    {
        const int waves = (MQ / 16) * (D_MODEL / 64);
        wmma_gemm_bias<<<dim3((waves + 7) / 8), 256, 0, stream>>>(
            query, W_off, b_off, off, MQ, D_MODEL, D_MODEL);
    }
    // 3) attention logits: query @ W_attn^T + b_attn -> [2400,128]
    {
        const int waves = (MQ / 16) * (128 / 64);
        wmma_gemm_bias<<<dim3((waves + 7) / 8), 256, 0, stream>>>(
            query, W_attn, b_attn, aw, MQ, 128, D_MODEL);
    }
    // 4) softmax over the 16 (level,point) logits per (b,q,h)
    {
        const int ngroups = MQ * N_HEADS;   // 19200
        softmax16_inplace<<<dim3((ngroups + 255) / 256), 256, 0, stream>>>(aw, ngroups);
    }
    // 5) bilinear sampling + weighted blend -> mid [2400,256]
    {
        const int waves = MQ * N_HEADS;     // 19200 waves, 8 per block
        ms_deform_sample<<<dim3((waves + 7) / 8), 256, 0, stream>>>(
            v_proj, off, aw, bbox, mid);
    }
    // 6) output projection: mid @ W_out^T + b_out -> d_out [2400,256]
    {
        const int waves = (MQ / 16) * (D_MODEL / 64);
        wmma_gemm_bias<<<dim3((waves + 7) / 8), 256, 0, stream>>>(
            mid, W_out, b_out, out, MQ, D_MODEL, D_MODEL);
    }
}